// HashEncoder_42932493091222
// MI455X (gfx1250) — compile-verified
//
#include <hip/hip_runtime.h>
#include <cmath>

// ---------------- Instant-NGP hash-grid constants ----------------
#define NLEVELS 16
#define TSIZE   (1u << 19)          // 2^19 entries per level
#define TMASK   (TSIZE - 1u)
#define PRIME1  2654435761u
#define PRIME2  805459861u
#define BLK     256                 // 8 wave32 waves per block

struct Scales { float s[NLEVELS]; };

typedef __attribute__((ext_vector_type(2))) float v2f;
typedef __attribute__((ext_vector_type(4))) int   v4i;

// Pointer types the gfx1250 async builtin expects (AS1 = __device__, AS3 = __shared__)
typedef v4i __attribute__((address_space(1)))* g_v4i_ptr;
typedef v4i __attribute__((address_space(3)))* l_v4i_ptr;

#if __has_builtin(__builtin_amdgcn_global_load_async_to_lds_b128)
#define HAVE_ASYNC_LDS 1
#else
#define HAVE_ASYNC_LDS 0
#if defined(__HIP_DEVICE_COMPILE__)
#warning "gfx1250 global_load_async_to_lds_b128 builtin NOT available in device pass"
#endif
#endif

__device__ __forceinline__ void wait_asynccnt0() {
#if __has_builtin(__builtin_amdgcn_s_wait_asynccnt)
    __builtin_amdgcn_s_wait_asynccnt(0);
#elif defined(__HIP_DEVICE_COMPILE__)
    asm volatile("s_wait_asynccnt 0" ::: "memory");
#endif
}

__global__ __launch_bounds__(BLK) void hashgrid_encode_kernel(
    const float* __restrict__ xg,   // [N,3]
    const float* __restrict__ Wg,   // [L,T,2]
    float* __restrict__ out,        // [N,L*2]
    int N, Scales sc) {

    const int tid = threadIdx.x;
    const long long base = (long long)blockIdx.x * BLK;

    float px, py, pz;

#if HAVE_ASYNC_LDS
    // ---- Stage this block's 3 KB slab of x into LDS with the CDNA5 async
    // global->LDS copy path: 192 lanes issue one coalesced b128 copy each. ----
    __shared__ float xs[BLK * 3];   // 3072 B = 192 * 16 B
    if (tid < (BLK * 3 * 4) / 16) {
        long long gbyte = base * 12 + (long long)tid * 16;
        const long long maxbyte = (long long)N * 12 - 16;
        if (gbyte > maxbyte) gbyte = maxbyte;   // tail clamp (data unused)
        void* gp = const_cast<char*>((const char*)xg) + gbyte;
        void* lp = (char*)xs + tid * 16;
        __builtin_amdgcn_global_load_async_to_lds_b128(
            (g_v4i_ptr)gp, (l_v4i_ptr)lp, 0, 0);
    }
    wait_asynccnt0();     // our wave's LDS writes are complete
    __syncthreads();      // everyone's writes visible
    px = xs[tid * 3 + 0];
    py = xs[tid * 3 + 1];
    pz = xs[tid * 3 + 2];
#else
    {
        long long i0 = base + tid;
        long long j = (i0 < N) ? i0 : (long long)N - 1;
        px = xg[j * 3 + 0];
        py = xg[j * 3 + 1];
        pz = xg[j * 3 + 2];
    }
#endif

    const long long i = base + tid;
    if (i >= N) return;

    const v2f* __restrict__ Wt = (const v2f*)Wg;      // [L*T] float2 entries
    float* __restrict__ op = out + (size_t)i * (NLEVELS * 2);

#pragma unroll
    for (int l = 0; l < NLEVELS; ++l) {
        const float s = sc.s[l];
        const float ux = px * s, uy = py * s, uz = pz * s;
        // x >= 0 so trunc == floor, matching astype(int32)
        const int ix = (int)ux, iy = (int)uy, iz = (int)uz;
        const float dx = ux - (float)ix;
        const float dy = uy - (float)iy;
        const float dz = uz - (float)iz;

        // int32-wraparound hash == uint32 arithmetic; % 2^19 == & TMASK
        const unsigned ha0 = (unsigned)ix;
        const unsigned ha1 = ha0 + 1u;
        const unsigned hb0 = (unsigned)iy * PRIME1;
        const unsigned hb1 = hb0 + PRIME1;
        const unsigned hc0 = (unsigned)iz * PRIME2;
        const unsigned hc1 = hc0 + PRIME2;
        const unsigned t00 = hb0 ^ hc0, t10 = hb1 ^ hc0;
        const unsigned t01 = hb0 ^ hc1, t11 = hb1 ^ hc1;

        const v2f* __restrict__ tl = Wt + (size_t)l * TSIZE;
        // corner v: bit0->x(+1), bit1->y(+1), bit2->z(+1)
        const v2f f0 = tl[(ha0 ^ t00) & TMASK];
        const v2f f1 = tl[(ha1 ^ t00) & TMASK];
        const v2f f2 = tl[(ha0 ^ t10) & TMASK];
        const v2f f3 = tl[(ha1 ^ t10) & TMASK];
        const v2f f4 = tl[(ha0 ^ t01) & TMASK];
        const v2f f5 = tl[(ha1 ^ t01) & TMASK];
        const v2f f6 = tl[(ha0 ^ t11) & TMASK];
        const v2f f7 = tl[(ha1 ^ t11) & TMASK];

        const float ox = 1.0f - dx, oy = 1.0f - dy, oz = 1.0f - dz;
        const float w00 = oy * oz, w10 = dy * oz;
        const float w01 = oy * dz, w11 = dy * dz;
        const float c0 = ox * w00, c1 = dx * w00;
        const float c2 = ox * w10, c3 = dx * w10;
        const float c4 = ox * w01, c5 = dx * w01;
        const float c6 = ox * w11, c7 = dx * w11;

        float r0 = f0.x * c0, r1 = f0.y * c0;
        r0 += f1.x * c1; r1 += f1.y * c1;
        r0 += f2.x * c2; r1 += f2.y * c2;
        r0 += f3.x * c3; r1 += f3.y * c3;
        r0 += f4.x * c4; r1 += f4.y * c4;
        r0 += f5.x * c5; r1 += f5.y * c5;
        r0 += f6.x * c6; r1 += f6.y * c6;
        r0 += f7.x * c7; r1 += f7.y * c7;

        // Non-temporal store: 256 MB output stream must NOT evict the
        // 64 MB L2-resident hash tables (L2 = 192 MB on MI455X).
        v2f r; r.x = r0; r.y = r1;
        __builtin_nontemporal_store(r, (v2f*)(op + l * 2));
    }
}

extern "C" void kernel_launch(void* const* d_in, const int* in_sizes, int n_in,
                              void* d_out, int out_size, void* d_ws, size_t ws_size,
                              hipStream_t stream) {
    const float* x = (const float*)d_in[0];   // [N,3] f32
    const float* W = (const float*)d_in[1];   // [16, 2^19, 2] f32
    float* out = (float*)d_out;               // [N,32] f32
    const int N = in_sizes[0] / 3;

    // Per-level scales, double precision like the reference, rounded to f32.
    Scales sc;
    const double b = exp((log(2048.0) - log(16.0)) / 16.0);
    for (int l = 0; l < NLEVELS; ++l)
        sc.s[l] = (float)(16.0 * pow(b, (double)l));

    const int blocks = (N + BLK - 1) / BLK;
    hipLaunchKernelGGL(hashgrid_encode_kernel, dim3(blocks), dim3(BLK), 0, stream,
                       x, W, out, N, sc);
}